// IntersectionGNN_11793980195028
// MI455X (gfx1250) — compile-verified
//
#include <hip/hip_runtime.h>
#include <hip/hip_bf16.h>

typedef __attribute__((ext_vector_type(2))) float v2f;
typedef __attribute__((ext_vector_type(8))) float v8f;

#define BATCH    4
#define N_NODES  50000
#define N_EDGES  800000
#define FEAT     64
#define N_LAYERS 3
#define M_TOTAL  (BATCH * N_NODES)          // 200000 rows
#define M_TILES  (M_TOTAL / 16)             // 12500
#define AGG_ELTS (M_TOTAL * FEAT)           // 12.8M elements
#define LDS_STRIDE 68                        // 16B-aligned rows, conflict-free A-frag reads

// Monotonic float->u32 key: float order == unsigned order.
//   key  = b ^ (((int)b >> 31) | 0x80000000)
//   b    = key ^ ((~(uint)((int)key >> 31)) | 0x80000000)
// key(-inf) = 0x007FFFFF  (segment_max identity / "empty" sentinel)
#define NEGINF_KEY 0x007FFFFFu

__device__ __forceinline__ unsigned f2key(unsigned b) {
    return b ^ (((unsigned)((int)b >> 31)) | 0x80000000u);
}
__device__ __forceinline__ float key2f(unsigned k) {
    unsigned xm = (~((unsigned)((int)k >> 31))) | 0x80000000u;
    return __uint_as_float(k ^ xm);
}

// ---------------------------------------------------------------------------
// 1) Fill agg keys with key(-inf). uint4 stores, exact grid (12500 x 256).
// ---------------------------------------------------------------------------
__global__ void fill_identity(uint4* __restrict__ agg) {
    int i = blockIdx.x * blockDim.x + threadIdx.x;   // 3.2M uint4
    agg[i] = make_uint4(NEGINF_KEY, NEGINF_KEY, NEGINF_KEY, NEGINF_KEY);
}

// ---------------------------------------------------------------------------
// 2) Scatter-max: thread = (batch, edge, feat). Lanes of a wave read a
//    contiguous 128B slice of the src row; branchless native u32 max atomic
//    (global_atomic_max_u32) on the order-preserving key.
// ---------------------------------------------------------------------------
__global__ void scatter_max(const unsigned* __restrict__ x,
                            const int* __restrict__ edges,
                            unsigned* __restrict__ agg) {
    int t = blockIdx.x * blockDim.x + threadIdx.x;   // 0 .. 51.2M-1 per batch
    int b = blockIdx.y;
    int e = t >> 6;
    int f = t & 63;
    int s = edges[e];                 // src
    int d = edges[N_EDGES + e];       // dst
    unsigned key = f2key(x[(b * N_NODES + s) * FEAT + f]);
    __hip_atomic_fetch_max(&agg[(b * N_NODES + d) * FEAT + f], key,
                           __ATOMIC_RELAXED, __HIP_MEMORY_SCOPE_AGENT);
}

// ---------------------------------------------------------------------------
// 3) Fused GEMM: out = decode(agg) @ Wr + br + xin @ Wroot
//    Block = 128 threads = 4 wave32. One 16-row M tile per block; wave w owns
//    N columns [16w, 16w+16). K = 64 -> 16 wmma_f32_16x16x4 steps per matrix.
//    Safe in-place (out == xin): xin rows are staged to LDS before stores.
// ---------------------------------------------------------------------------
__global__ void gemm_fused(const unsigned* __restrict__ agg,
                           const float* __restrict__ xin,
                           const float* __restrict__ Wr,
                           const float* __restrict__ br,
                           const float* __restrict__ Wroot,
                           float* __restrict__ out) {
    __shared__ float sA[16 * LDS_STRIDE];
    __shared__ float sX[16 * LDS_STRIDE];

    const int tid = threadIdx.x;
    const int m0  = blockIdx.x * 16;

    // Cooperative coalesced stage of the 16x64 tiles (8 elements per thread).
#pragma unroll
    for (int i = 0; i < 8; ++i) {
        int e = i * 128 + tid;
        int r = e >> 6;
        int c = e & 63;
        unsigned k = agg[(m0 + r) * FEAT + c];
        // empty segment (or true -inf) -> 0, matching jnp.where(isneginf, 0)
        sA[r * LDS_STRIDE + c] = (k == NEGINF_KEY) ? 0.0f : key2f(k);
        sX[r * LDS_STRIDE + c] = xin[(m0 + r) * FEAT + c];
    }
    __syncthreads();

    const int lane = tid & 31;
    const int wave = tid >> 5;
    const int n0   = wave * 16;
    const int mrow = lane & 15;       // A-frag: lane -> M
    const int half = lane >> 4;       // lanes 16-31 carry K+2,K+3
    const int n    = n0 + (lane & 15);

    float bv = br[n];
    v8f acc = {bv, bv, bv, bv, bv, bv, bv, bv};

    // acc += agg_tile @ Wr
#pragma unroll
    for (int ks = 0; ks < 16; ++ks) {
        int k = ks * 4 + 2 * half;
        v2f a, bm;
        a.x  = sA[mrow * LDS_STRIDE + k];
        a.y  = sA[mrow * LDS_STRIDE + k + 1];
        bm.x = Wr[(k)     * FEAT + n];
        bm.y = Wr[(k + 1) * FEAT + n];
        acc = __builtin_amdgcn_wmma_f32_16x16x4_f32(false, a, false, bm,
                                                    (short)0, acc, false, false);
    }
    // acc += x_tile @ Wroot
#pragma unroll
    for (int ks = 0; ks < 16; ++ks) {
        int k = ks * 4 + 2 * half;
        v2f a, bm;
        a.x  = sX[mrow * LDS_STRIDE + k];
        a.y  = sX[mrow * LDS_STRIDE + k + 1];
        bm.x = Wroot[(k)     * FEAT + n];
        bm.y = Wroot[(k + 1) * FEAT + n];
        acc = __builtin_amdgcn_wmma_f32_16x16x4_f32(false, a, false, bm,
                                                    (short)0, acc, false, false);
    }
    __syncthreads();

    // C/D layout: VGPR i, lanes 0-15 -> M=i, lanes 16-31 -> M=i+8
#pragma unroll
    for (int i = 0; i < 8; ++i) {
        out[(m0 + i + 8 * half) * FEAT + n] = acc[i];
    }
}

// ---------------------------------------------------------------------------
extern "C" void kernel_launch(void* const* d_in, const int* in_sizes, int n_in,
                              void* d_out, int out_size, void* d_ws, size_t ws_size,
                              hipStream_t stream) {
    const float* x     = (const float*)d_in[0];   // (4, 50000, 64) f32
    const int*   edges = (const int*)  d_in[1];   // (2, 800000) i32
    const float* Wrel  = (const float*)d_in[2];   // (3, 64, 64)
    const float* brel  = (const float*)d_in[3];   // (3, 64)
    const float* Wroot = (const float*)d_in[4];   // (3, 64, 64)
    float* out = (float*)d_out;                   // (4, 50000, 64)
    unsigned* agg = (unsigned*)d_ws;              // 51.2 MB scratch (keys)

    dim3 scat_grid(M_TOTAL, BATCH);               // 200000 x 4 blocks of 256

    for (int l = 0; l < N_LAYERS; ++l) {
        const float* xl = (l == 0) ? x : out;     // layers 1,2 run in-place on d_out
        fill_identity<<<AGG_ELTS / (4 * 256), 256, 0, stream>>>((uint4*)agg);
        scatter_max<<<scat_grid, 256, 0, stream>>>((const unsigned*)xl, edges, agg);
        gemm_fused<<<M_TILES, 128, 0, stream>>>(agg, xl,
                                                Wrel + l * FEAT * FEAT,
                                                brel + l * FEAT,
                                                Wroot + l * FEAT * FEAT,
                                                out);
    }
}